// EERNNSeqNet_15839839388008
// MI455X (gfx1250) — compile-verified
//
#include <hip/hip_runtime.h>
#include <hip/hip_bf16.h>
#include <math.h>

// ---------------------------------------------------------------------------
// EERNN seq-net for MI455X (gfx1250, wave32).
//   alpha GEMV  : HBM-bound streaming kernel (float4 loads, wave reduce)
//   top-k 64    : exact radix-select on monotonic float->u32 keys
//   attn + pred : single block, deterministic rank-sort + softmax
//   GRU cell    : v_wmma_f32_16x16x4_f32 matrix-core matvecs
// ---------------------------------------------------------------------------

typedef float v2f __attribute__((ext_vector_type(2)));
typedef float v8f __attribute__((ext_vector_type(8)));

#define QDIM 256
#define HDIM 256
#define TOPK 64

__device__ __forceinline__ unsigned keyOf(float f) {
    unsigned u = __float_as_uint(f);
    return (u & 0x80000000u) ? ~u : (u | 0x80000000u);   // order-preserving
}

// --------------------------- control-state reset ---------------------------
__global__ void zero_kernel(unsigned* hist, unsigned* state) {
    int t = threadIdx.x;
    hist[t]       = 0u;
    hist[256 + t] = 0u;
    hist[512 + t] = 0u;
    hist[768 + t] = 0u;
    if (t == 0) {
        state[0] = 0u;      // prefix
        state[1] = TOPK;    // still needed
        state[2] = 0u;      // cnt strictly-greater
        state[3] = 0u;      // cnt equal
    }
}

// ------------------- alpha = questions @ question (+hist0) -----------------
__global__ void alpha_kernel(const float* __restrict__ questions,
                             const float* __restrict__ question,
                             float* __restrict__ alpha,
                             unsigned* __restrict__ hist0, int n) {
    __shared__ unsigned lh[256];
    lh[threadIdx.x] = 0u;
    __syncthreads();

    const int lane   = threadIdx.x & 31;
    const int wavesB = blockDim.x >> 5;
    const int gwave  = blockIdx.x * wavesB + (threadIdx.x >> 5);
    const int nwaves = gridDim.x * wavesB;

    const float4* q4 = (const float4*)question;
    const float4 qa = q4[lane];
    const float4 qb = q4[lane + 32];

    for (int row = gwave; row < n; row += nwaves) {
        const float4* r4 = (const float4*)(questions + (size_t)row * QDIM);
        if (row + nwaves < n)   // keep HBM pipe full
            __builtin_prefetch(questions + (size_t)(row + nwaves) * QDIM + lane * 4, 0, 0);
        float4 a = r4[lane];
        float4 b = r4[lane + 32];
        float p = a.x * qa.x + a.y * qa.y + a.z * qa.z + a.w * qa.w
                + b.x * qb.x + b.y * qb.y + b.z * qb.z + b.w * qb.w;
        #pragma unroll
        for (int off = 16; off > 0; off >>= 1)
            p += __shfl_xor(p, off, 32);
        if (lane == 0) {
            alpha[row] = p;
            atomicAdd(&lh[keyOf(p) >> 24], 1u);
        }
    }
    __syncthreads();
    if (lh[threadIdx.x]) atomicAdd(&hist0[threadIdx.x], lh[threadIdx.x]);
}

// ------------------------- radix-select refinement -------------------------
__global__ void hist_kernel(const float* __restrict__ alpha, int n,
                            const unsigned* __restrict__ state,
                            unsigned* __restrict__ hist_base, int level) {
    __shared__ unsigned lh[256];
    lh[threadIdx.x] = 0u;
    __syncthreads();
    const unsigned P  = state[0];
    const int sh      = 32 - 8 * level;   // compare top 8*level bits
    const int binsh   = 24 - 8 * level;
    for (int i = blockIdx.x * blockDim.x + threadIdx.x; i < n;
         i += gridDim.x * blockDim.x) {
        unsigned u = keyOf(alpha[i]);
        if (((u ^ P) >> sh) == 0u)
            atomicAdd(&lh[(u >> binsh) & 0xFFu], 1u);
    }
    __syncthreads();
    if (lh[threadIdx.x])
        atomicAdd(&hist_base[256 * level + threadIdx.x], lh[threadIdx.x]);
}

__global__ void scan_kernel(const unsigned* __restrict__ hist_base,
                            unsigned* __restrict__ state, int level) {
    if (threadIdx.x != 0 || blockIdx.x != 0) return;
    const unsigned* h = hist_base + 256 * level;
    unsigned kneed = state[1];
    unsigned acc = 0u;
    int sel = 0;
    for (int b = 255; b >= 0; --b) {
        unsigned c = h[b];
        if (acc + c >= kneed) { sel = b; break; }
        acc += c;
    }
    state[0] |= ((unsigned)sel) << (24 - 8 * level);
    state[1]  = kneed - acc;     // ties still needed at threshold
}

// ------------------------------- selection ---------------------------------
__global__ void select_kernel(const float* __restrict__ alpha, int n,
                              unsigned* __restrict__ state,
                              float* __restrict__ sel_vals,
                              unsigned* __restrict__ sel_idx) {
    const unsigned T = state[0];
    for (int i = blockIdx.x * blockDim.x + threadIdx.x; i < n;
         i += gridDim.x * blockDim.x) {
        float v = alpha[i];
        unsigned u = keyOf(v);
        if (u > T) {
            unsigned pos = atomicAdd(&state[2], 1u);
            if (pos < TOPK) { sel_vals[pos] = v; sel_idx[pos] = (unsigned)i; }
        } else if (u == T) {
            unsigned pos = atomicAdd(&state[3], 1u);
            if (pos < TOPK) { sel_vals[TOPK + pos] = v; sel_idx[TOPK + pos] = (unsigned)i; }
        }
    }
}

// -------------------- softmax / attention / pred (1 block) -----------------
__global__ void fin_kernel(const float* __restrict__ question,
                           const float* __restrict__ hs,
                           const float* __restrict__ W_score,
                           const float* __restrict__ b_score,
                           const unsigned* __restrict__ state,
                           const float* __restrict__ sel_vals,
                           const unsigned* __restrict__ sel_idx,
                           float* __restrict__ out) {
    __shared__ float    v[TOPK];
    __shared__ unsigned id[TOPK];
    __shared__ float    sv[TOPK];
    __shared__ unsigned sid[TOPK];
    __shared__ float    w[TOPK];
    __shared__ float    red[256];
    const int t = threadIdx.x;

    unsigned G = state[2]; if (G > TOPK) G = TOPK;
    unsigned E = state[3]; if (E > TOPK) E = TOPK;
    if (t < TOPK) {
        if ((unsigned)t < G) { v[t] = sel_vals[t]; id[t] = sel_idx[t]; }
        else {
            unsigned e = (unsigned)t - G;
            if (e < E) { v[t] = sel_vals[TOPK + e]; id[t] = sel_idx[TOPK + e]; }
            else       { v[t] = -3.0e38f;           id[t] = 0u; }
        }
    }
    __syncthreads();
    // deterministic rank sort (value desc, index asc) -> fixed accumulation order
    if (t < TOPK) {
        float vt = v[t]; unsigned it = id[t];
        int rank = 0;
        for (int j = 0; j < TOPK; ++j) {
            float vj = v[j]; unsigned ij = id[j];
            if (vj > vt || (vj == vt && (ij < it || (ij == it && j < t)))) rank++;
        }
        sv[rank] = vt; sid[rank] = it;
    }
    __syncthreads();
    if (t == 0) {
        float m = sv[0];
        for (int k = 1; k < TOPK; ++k) m = fmaxf(m, sv[k]);
        float s = 0.f;
        for (int k = 0; k < TOPK; ++k) { w[k] = expf(sv[k] - m); s += w[k]; }
        float inv = 1.0f / s;
        for (int k = 0; k < TOPK; ++k) w[k] *= inv;
    }
    __syncthreads();
    float ah = 0.f;
    for (int k = 0; k < TOPK; ++k)
        ah += w[k] * hs[(size_t)sid[k] * HDIM + t];     // coalesced per k
    red[t] = W_score[t] * question[t] + W_score[QDIM + t] * ah;
    __syncthreads();
    for (int s2 = 128; s2 > 0; s2 >>= 1) {
        if (t < s2) red[t] += red[t + s2];
        __syncthreads();
    }
    if (t == 0) out[0] = red[0] + b_score[0];
}

// --------------------------- GRU cell via WMMA -----------------------------
// One block = 8 waves. 96 row-tiles of 16 (48 for W_ih K=512, 48 for W_hh
// K=256). A fragment: 16x4 f32 tile of W per ISA layout (lanes 0-15: K=0,1;
// lanes 16-31: K=2,3). B replicates the x/h slice across all 16 columns so
// every D column holds the same matvec result; extract column 0 from the
// documented C/D layout (row m -> VGPR m @ lane 0, VGPR m-8 @ lane 16).
__global__ void gru_kernel(const float* __restrict__ question,
                           const float* __restrict__ score,
                           const float* __restrict__ hs, int n,
                           const float* __restrict__ W_ih,
                           const float* __restrict__ W_hh,
                           const float* __restrict__ b_ih,
                           const float* __restrict__ b_hh,
                           float* __restrict__ out) {
    __shared__ float xb[2 * QDIM];
    __shared__ float hb[HDIM];
    __shared__ float g[6 * HDIM];    // gx[768] | gh[768]
    const int t = threadIdx.x;
    {
        float q = question[t];
        float right = (score[0] >= 0.5f) ? 1.0f : 0.0f;
        xb[t]        = q * right;
        xb[QDIM + t] = q * (1.0f - right);
        hb[t] = hs[(size_t)(n - 1) * HDIM + t];
    }
    __syncthreads();

    const int lane = t & 31;
    const int wave = t >> 5;
    const int m    = lane & 15;
    const int koff = (lane >> 4) << 1;   // 0 for lanes 0-15, 2 for 16-31

    for (int tile = wave; tile < 96; tile += 8) {      // uniform: EXEC all-1s
        const float* W; const float* src; float* dst; int ld, K, r0;
        if (tile < 48) { W = W_ih; src = xb; dst = g;       ld = 512; K = 512; r0 = tile * 16; }
        else           { W = W_hh; src = hb; dst = g + 768; ld = 256; K = 256; r0 = (tile - 48) * 16; }

        v8f acc = {0.f, 0.f, 0.f, 0.f, 0.f, 0.f, 0.f, 0.f};
        const float* wrow = W + (size_t)(r0 + m) * ld;
        for (int k = 0; k < K; k += 4) {
            v2f a, b;
            a.x = wrow[k + koff];
            a.y = wrow[k + koff + 1];
            b.x = src[k + koff];
            b.y = src[k + koff + 1];
            acc = __builtin_amdgcn_wmma_f32_16x16x4_f32(
                false, a, false, b, (short)0, acc, false, false);
        }
        if (lane == 0) {
            #pragma unroll
            for (int vv = 0; vv < 8; ++vv) dst[r0 + vv] = acc[vv];
        } else if (lane == 16) {
            #pragma unroll
            for (int vv = 0; vv < 8; ++vv) dst[r0 + 8 + vv] = acc[vv];
        }
    }
    __syncthreads();
    {
        float xr = g[t]        + b_ih[t];
        float xz = g[256 + t]  + b_ih[256 + t];
        float xn = g[512 + t]  + b_ih[512 + t];
        float hr = g[768 + t]  + b_hh[t];
        float hz = g[1024 + t] + b_hh[256 + t];
        float hn = g[1280 + t] + b_hh[512 + t];
        float r  = 1.0f / (1.0f + expf(-(xr + hr)));
        float z  = 1.0f / (1.0f + expf(-(xz + hz)));
        float nn = tanhf(xn + r * hn);
        out[1 + t] = (1.0f - z) * nn + z * hb[t];
    }
}

// ------------------------------ launcher -----------------------------------
extern "C" void kernel_launch(void* const* d_in, const int* in_sizes, int n_in,
                              void* d_out, int out_size, void* d_ws, size_t ws_size,
                              hipStream_t stream) {
    const float* question  = (const float*)d_in[0];
    const float* score     = (const float*)d_in[1];
    const float* questions = (const float*)d_in[2];
    const float* hs        = (const float*)d_in[3];
    const float* W_score   = (const float*)d_in[4];
    const float* b_score   = (const float*)d_in[5];
    const float* W_ih      = (const float*)d_in[6];
    const float* W_hh      = (const float*)d_in[7];
    const float* b_ih      = (const float*)d_in[8];
    const float* b_hh      = (const float*)d_in[9];
    float* out = (float*)d_out;

    const int n = in_sizes[2] / QDIM;     // 1,000,000 rows

    // workspace layout
    size_t off = ((size_t)n * sizeof(float) + 255) & ~(size_t)255;
    float*    alpha    = (float*)d_ws;
    unsigned* hist     = (unsigned*)((char*)d_ws + off);      // 4 x 256 bins
    unsigned* state    = hist + 1024;                         // 8 u32
    float*    sel_vals = (float*)(state + 8);                 // 128 floats
    unsigned* sel_idx  = (unsigned*)(sel_vals + 128);         // 128 u32

    zero_kernel<<<1, 256, 0, stream>>>(hist, state);
    alpha_kernel<<<4096, 256, 0, stream>>>(questions, question, alpha, hist, n);
    scan_kernel<<<1, 1, 0, stream>>>(hist, state, 0);
    hist_kernel<<<2048, 256, 0, stream>>>(alpha, n, state, hist, 1);
    scan_kernel<<<1, 1, 0, stream>>>(hist, state, 1);
    hist_kernel<<<2048, 256, 0, stream>>>(alpha, n, state, hist, 2);
    scan_kernel<<<1, 1, 0, stream>>>(hist, state, 2);
    hist_kernel<<<2048, 256, 0, stream>>>(alpha, n, state, hist, 3);
    scan_kernel<<<1, 1, 0, stream>>>(hist, state, 3);
    select_kernel<<<2048, 256, 0, stream>>>(alpha, n, state, sel_vals, sel_idx);
    fin_kernel<<<1, 256, 0, stream>>>(question, hs, W_score, b_score,
                                      state, sel_vals, sel_idx, out);
    gru_kernel<<<1, 256, 0, stream>>>(question, score, hs, n,
                                      W_ih, W_hh, b_ih, b_hh, out);
}